// GAT_Full_17016660426788
// MI455X (gfx1250) — compile-verified
//
#include <hip/hip_runtime.h>

typedef __attribute__((ext_vector_type(2))) float v2f;
typedef __attribute__((ext_vector_type(8))) float v8f;

#define HEADS 4
#define DHEAD 32
#define FDIM 128           // HEADS * DHEAD
#define NEG_SLOPE 0.2f

__device__ __forceinline__ float leaky(float v) {
  return v > 0.f ? v : NEG_SLOPE * v;
}

__device__ __forceinline__ void atomicMaxFloat(float* addr, float val) {
  // Works with init = -inf. Positive values: signed-int max; negative: unsigned min.
  if (val >= 0.f) {
    atomicMax((int*)addr, __float_as_int(val));
  } else {
    atomicMin((unsigned int*)addr, __float_as_uint(val));
  }
}

__device__ __forceinline__ void atomicAddF32(float* addr, float val) {
  __hip_atomic_fetch_add(addr, val, __ATOMIC_RELAXED, __HIP_MEMORY_SCOPE_AGENT);
}

// ---------------------------------------------------------------------------
// GEMM: C[n,128] = A[n,128] x W[128,128], fp32 WMMA 16x16x4.
// One block = 16 rows; 8 waves, wave w owns output columns [16w, 16w+16).
// A strip staged in LDS (padded stride 132 -> conflict-free fragment reads).
// B fragments read from global (W = 64KB, L2/L0 resident across all blocks).
// ---------------------------------------------------------------------------
__global__ __launch_bounds__(256) void gemm128_wmma(const float* __restrict__ A,
                                                    const float* __restrict__ W,
                                                    float* __restrict__ C,
                                                    int nrows) {
  __shared__ float As[16][132];
  const int tid  = threadIdx.x;
  const int row0 = blockIdx.x * 16;

  // Cooperative load of the 16x128 A strip (coalesced float4, 2 per thread).
  for (int i = tid; i < 16 * 32; i += 256) {
    const int r  = i >> 5;
    const int c  = (i & 31) << 2;
    const int gr = row0 + r;
    float4 av = make_float4(0.f, 0.f, 0.f, 0.f);
    if (gr < nrows) av = *(const float4*)(A + (size_t)gr * FDIM + c);
    As[r][c + 0] = av.x; As[r][c + 1] = av.y;
    As[r][c + 2] = av.z; As[r][c + 3] = av.w;
  }
  __syncthreads();

  const int lane = tid & 31;
  const int wv   = tid >> 5;       // wave id -> 16-column tile
  const int m16  = lane & 15;
  const int half = lane >> 4;
  const int col  = wv * 16 + m16;

  v8f acc = {};
#pragma unroll
  for (int kk = 0; kk < FDIM; kk += 4) {
    const int ka = kk + 2 * half;
    // A fragment: lane holds A[m16][ka], A[m16][ka+1]
    v2f a; a.x = As[m16][ka]; a.y = As[m16][ka + 1];
    // B fragment: lane holds W[ka][col], W[ka+1][col]
    v2f b; b.x = W[(size_t)ka * FDIM + col];
    b.y = W[(size_t)(ka + 1) * FDIM + col];
    acc = __builtin_amdgcn_wmma_f32_16x16x4_f32(
        /*neg_a=*/false, a, /*neg_b=*/false, b,
        /*c_mod=*/(short)0, acc, /*reuse_a=*/false, /*reuse_b=*/false);
  }

  // C/D layout: VGPR v -> row (v + 8*half), col = lane&15 within tile.
#pragma unroll
  for (int v = 0; v < 8; ++v) {
    const int gr = row0 + v + 8 * half;
    if (gr < nrows) C[(size_t)gr * FDIM + col] = acc[v];
  }
}

// ---------------------------------------------------------------------------
// Per-(node,head) attention scores el/er; also (re)initializes m = -inf,
// denom = 0. One wave32 per (node,head); lane d handles feature d (D=32).
// ---------------------------------------------------------------------------
__global__ __launch_bounds__(256) void node_scores(const float* __restrict__ feat,
                                                   const float* __restrict__ al,
                                                   const float* __restrict__ ar,
                                                   float* __restrict__ el,
                                                   float* __restrict__ er,
                                                   float* __restrict__ mbuf,
                                                   float* __restrict__ denom,
                                                   int n) {
  const int lane  = threadIdx.x & 31;
  const long pair = (long)blockIdx.x * 8 + (threadIdx.x >> 5);
  if (pair >= (long)n * HEADS) return;
  const int node = (int)(pair / HEADS);
  const int h    = (int)(pair % HEADS);
  const float f  = feat[(size_t)node * FDIM + h * DHEAD + lane];
  float vl = f * al[h * DHEAD + lane];
  float vr = f * ar[h * DHEAD + lane];
#pragma unroll
  for (int off = 16; off; off >>= 1) {
    vl += __shfl_down(vl, off);
    vr += __shfl_down(vr, off);
  }
  if (lane == 0) {
    el[pair]    = vl;
    er[pair]    = vr;
    mbuf[pair]  = -__builtin_inff();
    denom[pair] = 0.f;
  }
}

// ---------------------------------------------------------------------------
// Segment max over incoming edges; one THREAD per edge, all 4 heads via
// float4 (b128) loads of el[src]/er[dst], 4 atomics into one cacheline.
// ---------------------------------------------------------------------------
__global__ __launch_bounds__(256) void edge_max4(const float4* __restrict__ el4,
                                                 const float4* __restrict__ er4,
                                                 const int* __restrict__ src,
                                                 const int* __restrict__ dst,
                                                 float* __restrict__ mbuf,
                                                 int ne) {
  const int i = blockIdx.x * 256 + threadIdx.x;
  if (i >= ne) return;
  const int s = src[i], t = dst[i];
  const float4 l = el4[s];
  const float4 r = er4[t];
  float* mb = mbuf + (size_t)t * HEADS;
  atomicMaxFloat(mb + 0, leaky(l.x + r.x));
  atomicMaxFloat(mb + 1, leaky(l.y + r.y));
  atomicMaxFloat(mb + 2, leaky(l.z + r.z));
  atomicMaxFloat(mb + 3, leaky(l.w + r.w));
}

// Segment sum of exp(e - m[dst]); one thread per edge, float4 loads.
__global__ __launch_bounds__(256) void edge_expsum4(const float4* __restrict__ el4,
                                                    const float4* __restrict__ er4,
                                                    const float4* __restrict__ m4,
                                                    const int* __restrict__ src,
                                                    const int* __restrict__ dst,
                                                    float* __restrict__ denom,
                                                    int ne) {
  const int i = blockIdx.x * 256 + threadIdx.x;
  if (i >= ne) return;
  const int s = src[i], t = dst[i];
  const float4 l = el4[s];
  const float4 r = er4[t];
  const float4 mm = m4[t];
  float* dn = denom + (size_t)t * HEADS;
  atomicAddF32(dn + 0, __expf(leaky(l.x + r.x) - mm.x));
  atomicAddF32(dn + 1, __expf(leaky(l.y + r.y) - mm.y));
  atomicAddF32(dn + 2, __expf(leaky(l.z + r.z) - mm.z));
  atomicAddF32(dn + 3, __expf(leaky(l.w + r.w) - mm.w));
}

// ---------------------------------------------------------------------------
// Weighted scatter-aggregate. One WAVE per edge: lane c owns features
// [4c, 4c+4) of the full [H=4, D=32] row via one b128 load; the head
// coefficient a[h] is selected by c>>3. Atomic f32 adds do the scatter.
// mode 0: out[dst, h, d] += a_h * feat[src, h, d]                ([N,128])
// mode 1: out[dst, d]    += scale * a_h * feat[src, h, d]        ([N,32], head-mean)
// ---------------------------------------------------------------------------
__global__ __launch_bounds__(256) void edge_aggr4(const float4* __restrict__ feat4,
                                                  const float4* __restrict__ el4,
                                                  const float4* __restrict__ er4,
                                                  const float4* __restrict__ m4,
                                                  const float4* __restrict__ d4,
                                                  const int* __restrict__ src,
                                                  const int* __restrict__ dst,
                                                  float* __restrict__ out,
                                                  int ne, float scale, int mode) {
  const int  lane = threadIdx.x & 31;
  const long e    = (long)blockIdx.x * 8 + (threadIdx.x >> 5);
  if (e >= ne) return;
  const int s = src[e], t = dst[e];
  const float4 l  = el4[s];
  const float4 r  = er4[t];
  const float4 mm = m4[t];
  const float4 dd = d4[t];
  const float a0 = scale * __expf(leaky(l.x + r.x) - mm.x) / dd.x;
  const float a1 = scale * __expf(leaky(l.y + r.y) - mm.y) / dd.y;
  const float a2 = scale * __expf(leaky(l.z + r.z) - mm.z) / dd.z;
  const float a3 = scale * __expf(leaky(l.w + r.w) - mm.w) / dd.w;

  const int h = lane >> 3;                       // 8 lanes per head
  const float a = (h == 0) ? a0 : (h == 1) ? a1 : (h == 2) ? a2 : a3;

  const float4 f = feat4[(size_t)s * (FDIM / 4) + lane];   // 512B/wave, coalesced
  if (mode == 0) {
    float* o = out + (size_t)t * FDIM + lane * 4;
    atomicAddF32(o + 0, a * f.x);
    atomicAddF32(o + 1, a * f.y);
    atomicAddF32(o + 2, a * f.z);
    atomicAddF32(o + 3, a * f.w);
  } else {
    float* o = out + (size_t)t * DHEAD + ((lane * 4) & (DHEAD - 1));
    atomicAddF32(o + 0, a * f.x);
    atomicAddF32(o + 1, a * f.y);
    atomicAddF32(o + 2, a * f.z);
    atomicAddF32(o + 3, a * f.w);
  }
}

// h = relu(rst + b1), flattened [N,128].
__global__ __launch_bounds__(256) void epilogue1(const float* __restrict__ rst,
                                                 const float* __restrict__ b1,
                                                 float* __restrict__ hbuf,
                                                 long total) {
  const long i = (long)blockIdx.x * blockDim.x + threadIdx.x;
  if (i >= total) return;
  const int c = (int)(i & (FDIM - 1));
  const float v = rst[i] + b1[c];
  hbuf[i] = v > 0.f ? v : 0.f;
}

// d_out init for layer 2: out[n,d] = (1/H) * sum_h (h[n,h,d] + b2[h,d])
// (residual + bias, head-mean); aggregation then atomically adds (1/H)*a*feat.
__global__ __launch_bounds__(256) void out_init(const float* __restrict__ hbuf,
                                                const float* __restrict__ b2,
                                                float* __restrict__ out,
                                                long total) {
  const long i = (long)blockIdx.x * blockDim.x + threadIdx.x;
  if (i >= total) return;
  const int node = (int)(i >> 5);        // DHEAD == 32
  const int d    = (int)(i & 31);
  float s = 0.f;
#pragma unroll
  for (int hh = 0; hh < HEADS; ++hh)
    s += hbuf[(size_t)node * FDIM + hh * DHEAD + d] + b2[hh * DHEAD + d];
  out[i] = s * (1.f / HEADS);
}

extern "C" void kernel_launch(void* const* d_in, const int* in_sizes, int n_in,
                              void* d_out, int out_size, void* d_ws, size_t ws_size,
                              hipStream_t stream) {
  const float* x   = (const float*)d_in[0];
  const int*   src = (const int*)d_in[1];
  const int*   dst = (const int*)d_in[2];
  const float* W1  = (const float*)d_in[3];
  const float* al1 = (const float*)d_in[4];
  const float* ar1 = (const float*)d_in[5];
  const float* b1  = (const float*)d_in[6];
  const float* W2  = (const float*)d_in[7];
  const float* al2 = (const float*)d_in[8];
  const float* ar2 = (const float*)d_in[9];
  const float* b2  = (const float*)d_in[10];
  float* out = (float*)d_out;

  const int n = in_sizes[0] / FDIM;   // 50000
  const int e = in_sizes[1];          // 800000

  // Workspace layout (floats): F[n*128] | R[n*128] | el,er,m,denom [n*H each]
  // All segments are 16B-aligned (n*H*4 bytes is a multiple of 16).
  float* F  = (float*)d_ws;
  float* R  = F  + (size_t)n * FDIM;
  float* el = R  + (size_t)n * FDIM;
  float* er = el + (size_t)n * HEADS;
  float* mx = er + (size_t)n * HEADS;
  float* dn = mx + (size_t)n * HEADS;
  const float4* el4 = (const float4*)el;
  const float4* er4 = (const float4*)er;
  const float4* mx4 = (const float4*)mx;
  const float4* dn4 = (const float4*)dn;

  const long nhPairs = (long)n * HEADS;
  const int  gemmBlocks = (n + 15) / 16;
  const int  nsBlocks   = (int)((nhPairs + 7) / 8);
  const int  eBlocks    = (e + 255) / 256;
  const int  agBlocks   = (e + 7) / 8;

  // ---------------- Layer 1 ----------------
  gemm128_wmma<<<gemmBlocks, 256, 0, stream>>>(x, W1, F, n);             // F = feat1
  node_scores<<<nsBlocks, 256, 0, stream>>>(F, al1, ar1, el, er, mx, dn, n);
  edge_max4<<<eBlocks, 256, 0, stream>>>(el4, er4, src, dst, mx, e);
  edge_expsum4<<<eBlocks, 256, 0, stream>>>(el4, er4, mx4, src, dst, dn, e);
  hipMemsetAsync(R, 0, (size_t)n * FDIM * sizeof(float), stream);        // R = rst1 = 0
  edge_aggr4<<<agBlocks, 256, 0, stream>>>((const float4*)F, el4, er4, mx4, dn4,
                                           src, dst, R, e, 1.0f, /*mode=*/0);
  epilogue1<<<(int)(((long)n * FDIM + 255) / 256), 256, 0, stream>>>(R, b1, F,
                                                                     (long)n * FDIM);
  // F now holds h (layer-2 input / residual)

  // ---------------- Layer 2 ----------------
  gemm128_wmma<<<gemmBlocks, 256, 0, stream>>>(F, W2, R, n);             // R = feat2
  node_scores<<<nsBlocks, 256, 0, stream>>>(R, al2, ar2, el, er, mx, dn, n);
  edge_max4<<<eBlocks, 256, 0, stream>>>(el4, er4, src, dst, mx, e);
  edge_expsum4<<<eBlocks, 256, 0, stream>>>(el4, er4, mx4, src, dst, dn, e);
  out_init<<<(int)(((long)n * DHEAD + 255) / 256), 256, 0, stream>>>(F, b2, out,
                                                                     (long)n * DHEAD);
  edge_aggr4<<<agBlocks, 256, 0, stream>>>((const float4*)R, el4, er4, mx4, dn4,
                                           src, dst, out, e, 1.0f / HEADS, /*mode=*/1);
}